// DistanceLoss_11158325035367
// MI455X (gfx1250) — compile-verified
//
#include <hip/hip_runtime.h>
#include <hip/hip_bf16.h>

typedef __attribute__((ext_vector_type(2))) float v2f;
typedef __attribute__((ext_vector_type(8))) float v8f;

#define PIXELS      102400   // 320*320
#define NUM_LABELS  64
#define PER_SAMPLE  384      // 4*64 ksum (channel-major) + 64 kcnt + 64 tcnt
#define CHUNK       4096     // pixels per block (25 blocks per sample)
#define DELTA_DIS   3.0f

// ---------------------------------------------------------------------------
// Kernel 0: zero the workspace accumulators (d_ws is poisoned by the harness)
// ---------------------------------------------------------------------------
__global__ void dl_zero_kernel(float* __restrict__ ws, int count) {
    int i = blockIdx.x * blockDim.x + threadIdx.x;
    if (i < count) ws[i] = 0.0f;
}

// ---------------------------------------------------------------------------
// Kernel 1: bandwidth-bound segment reduction.
// LDS-privatized per-block histogram (ds_add_f32), then one global atomic
// add per non-zero bin into ws[sample*384 + bin].
// Layout per sample: [c*64 + l] ksum (c=0..3), [256+l] kcnt, [320+l] tcnt.
// ---------------------------------------------------------------------------
__global__ void dl_accum_kernel(const float* __restrict__ preds,
                                const int*   __restrict__ targets,
                                float* __restrict__ ws) {
    const int n     = blockIdx.y;            // sample
    const int chunk = blockIdx.x;            // 0..24
    const int tid   = threadIdx.x;           // 0..255

    __shared__ float h[PER_SAMPLE];
    for (int i = tid; i < PER_SAMPLE; i += blockDim.x) h[i] = 0.0f;
    __syncthreads();

    const float* sim0 = preds   + ((size_t)n * 6 + 2) * PIXELS;  // channels 2..5
    const int*   text = targets + ((size_t)n * 2 + 0) * PIXELS;
    const int*   kern = targets + ((size_t)n * 2 + 1) * PIXELS;

    const int start = chunk * CHUNK;
    #pragma unroll 4
    for (int p = start + tid; p < start + CHUNK; p += blockDim.x) {
        const int t = text[p] & (NUM_LABELS - 1);
        const int k = kern[p] & (NUM_LABELS - 1);
        const float s0 = sim0[p];
        const float s1 = sim0[p + PIXELS];
        const float s2 = sim0[p + 2 * PIXELS];
        const float s3 = sim0[p + 3 * PIXELS];
        // LDS float atomics -> ds_add_f32 (workgroup scope)
        __hip_atomic_fetch_add(&h[0 * 64 + k], s0,  __ATOMIC_RELAXED, __HIP_MEMORY_SCOPE_WORKGROUP);
        __hip_atomic_fetch_add(&h[1 * 64 + k], s1,  __ATOMIC_RELAXED, __HIP_MEMORY_SCOPE_WORKGROUP);
        __hip_atomic_fetch_add(&h[2 * 64 + k], s2,  __ATOMIC_RELAXED, __HIP_MEMORY_SCOPE_WORKGROUP);
        __hip_atomic_fetch_add(&h[3 * 64 + k], s3,  __ATOMIC_RELAXED, __HIP_MEMORY_SCOPE_WORKGROUP);
        __hip_atomic_fetch_add(&h[256 + k],    1.0f, __ATOMIC_RELAXED, __HIP_MEMORY_SCOPE_WORKGROUP);
        __hip_atomic_fetch_add(&h[320 + t],    1.0f, __ATOMIC_RELAXED, __HIP_MEMORY_SCOPE_WORKGROUP);
    }
    __syncthreads();

    float* dst = ws + (size_t)n * PER_SAMPLE;
    for (int i = tid; i < PER_SAMPLE; i += blockDim.x) {
        const float v = h[i];
        if (v != 0.0f)
            __hip_atomic_fetch_add(&dst[i], v, __ATOMIC_RELAXED, __HIP_MEMORY_SCOPE_AGENT);
    }
}

// ---------------------------------------------------------------------------
// Kernel 2: one wave (32 threads) per sample.
// Gram = G*G^T (64x64, K=4) via 10 upper-triangular V_WMMA_F32_16X16X4_F32
// tiles; ||Gi-Gj||^2 = ni + nj - 2*Gram[i][j]; masked pair loss; reduce.
// ---------------------------------------------------------------------------
__global__ void dl_finalize_kernel(const float* __restrict__ ws,
                                   float* __restrict__ out) {
    const int n    = blockIdx.x;
    const int lane = threadIdx.x;        // 0..31, single wave32
    const int m    = lane & 15;
    const int half = lane >> 4;

    const float* base = ws + (size_t)n * PER_SAMPLE;

    __shared__ float Gs[4][NUM_LABELS + 1];   // channel-major, padded row
    __shared__ float norms[NUM_LABELS];
    __shared__ int   validsh[NUM_LABELS];
    __shared__ float partial[32];

    for (int l = lane; l < NUM_LABELS; l += 32) {
        const float kc = base[256 + l];
        const float tc = base[320 + l];
        const int   valid = (l >= 1) && (kc > 0.0f) && (tc > 0.0f);
        const float denom = fmaxf(kc, 1.0f);
        const float g0 = base[0 * 64 + l] / denom;   // IEEE divide, matches reference
        const float g1 = base[1 * 64 + l] / denom;
        const float g2 = base[2 * 64 + l] / denom;
        const float g3 = base[3 * 64 + l] / denom;
        Gs[0][l] = g0; Gs[1][l] = g1; Gs[2][l] = g2; Gs[3][l] = g3;
        norms[l]   = g0 * g0 + g1 * g1 + g2 * g2 + g3 * g3;
        validsh[l] = valid;
    }
    __syncthreads();

    float total = 0.0f;
    #pragma unroll
    for (int mi = 0; mi < 4; ++mi) {
        // A tile: rows mi*16..mi*16+15 of G; lane half selects K pair {0,1}/{2,3}
        v2f a;
        a.x = Gs[2 * half + 0][mi * 16 + m];
        a.y = Gs[2 * half + 1][mi * 16 + m];
        #pragma unroll
        for (int ni = mi; ni < 4; ++ni) {
            v2f b;
            b.x = Gs[2 * half + 0][ni * 16 + m];
            b.y = Gs[2 * half + 1][ni * 16 + m];
            v8f acc = {};
            // D = A(16x4) x B(4x16) + C : Gram tile in f32 matrix pipe
            acc = __builtin_amdgcn_wmma_f32_16x16x4_f32(
                /*neg_a=*/false, a, /*neg_b=*/false, b,
                /*c_mod=*/(short)0, acc, /*reuse_a=*/false, /*reuse_b=*/false);
            #pragma unroll
            for (int r = 0; r < 8; ++r) {
                const int i = mi * 16 + r + 8 * half;   // C/D layout: row=r+8*(lane/16)
                const int j = ni * 16 + m;              // col = lane%16
                if (j > i && validsh[i] && validsh[j]) {
                    float sq = norms[i] + norms[j] - 2.0f * acc[r];
                    sq = fmaxf(sq, 0.0f);
                    const float D = sqrtf(sq);
                    const float t = fmaxf(DELTA_DIS - D, 0.0f);
                    total += logf(t * t + 1.0f);
                }
            }
        }
    }

    partial[lane] = total;
    __syncthreads();
    if (lane == 0) {
        float s = 0.0f;
        #pragma unroll
        for (int i = 0; i < 32; ++i) s += partial[i];
        float Nv = 0.0f;
        for (int l = 0; l < NUM_LABELS; ++l) Nv += (float)validsh[l];
        out[n] = (Nv > 1.0f) ? (s / (Nv * (Nv - 1.0f))) : 0.0f;
    }
}

// ---------------------------------------------------------------------------
extern "C" void kernel_launch(void* const* d_in, const int* in_sizes, int n_in,
                              void* d_out, int out_size, void* d_ws, size_t ws_size,
                              hipStream_t stream) {
    const float* preds   = (const float*)d_in[0];   // [32, 6, 320, 320] f32
    const int*   targets = (const int*)d_in[1];     // [32, 2, 320, 320] i32
    float*       out     = (float*)d_out;           // [32] f32
    float*       ws      = (float*)d_ws;

    const int num_samples = in_sizes[0] / (6 * PIXELS);   // 32
    const int ws_count    = num_samples * PER_SAMPLE;     // 12288 floats

    dl_zero_kernel<<<(ws_count + 255) / 256, 256, 0, stream>>>(ws, ws_count);

    dim3 grid(PIXELS / CHUNK, num_samples);               // (25, 32)
    dl_accum_kernel<<<grid, 256, 0, stream>>>(preds, targets, ws);

    dl_finalize_kernel<<<num_samples, 32, 0, stream>>>(ws, out);
}